// GCN_24867860644026
// MI455X (gfx1250) — compile-verified
//
#include <hip/hip_runtime.h>

#define D_IN 64
#define TILE 16

typedef __attribute__((ext_vector_type(2))) float v2f;
typedef __attribute__((ext_vector_type(8))) float v8f;

// ---------------------------------------------------------------------------
// Kernel 1: zero the aggregation buffer (float4 stores, grid-stride)
// ---------------------------------------------------------------------------
__global__ void gcn_zero_kernel(float4* __restrict__ p, int n4) {
    int i = blockIdx.x * blockDim.x + threadIdx.x;
    int stride = gridDim.x * blockDim.x;
    float4 z = make_float4(0.f, 0.f, 0.f, 0.f);
    for (; i < n4; i += stride) p[i] = z;
}

// ---------------------------------------------------------------------------
// Kernel 2: weighted gather + scatter-add.
// One wave32 per edge; each lane handles 2 consecutive feature dims via a
// coalesced float2 load, then two global fp32 atomic adds into agg[dst].
// agg (25.6 MB) is L2-resident on MI455X (192 MB L2), so the atomics
// resolve at L2 atomic throughput.
// ---------------------------------------------------------------------------
__global__ void gcn_scatter_kernel(const int* __restrict__ src,
                                   const int* __restrict__ dst,
                                   const float* __restrict__ w,
                                   const float* __restrict__ data,
                                   float* __restrict__ agg,
                                   int n_edges) {
    int gid  = blockIdx.x * blockDim.x + threadIdx.x;
    int e    = gid >> 5;        // one wave32 per edge
    int lane = gid & 31;        // lane handles dims [2*lane, 2*lane+1]
    if (e >= n_edges) return;

    int   s  = src[e];
    int   d  = dst[e];
    float wt = w[e];

    const float2* dp = reinterpret_cast<const float2*>(data + (size_t)s * D_IN);
    float2 v = dp[lane];

    float* out = agg + (size_t)d * D_IN + lane * 2;
    atomicAdd(out,     v.x * wt);
    atomicAdd(out + 1, v.y * wt);
}

// ---------------------------------------------------------------------------
// Kernel 3: out[N,64] = agg[N,64] @ theta[64,64] using V_WMMA_F32_16X16X4_F32.
// One wave per 16x16 output tile; K=64 consumed in 16 chained WMMA steps.
//
// f32 16x16x4 A layout (ISA 7.12.2): lanes 0-15 = rows M=0..15 with
//   VGPR0=K0, VGPR1=K1; lanes 16-31 = same rows with K2,K3.
//   -> per lane a contiguous float2 at (rowBase + (lane&15), kk*4 + 2*(lane>>4)).
// B (4x16, row striped across lanes): lane (lane&15)=col; VGPR0 holds row
//   k0, VGPR1 holds row k0+1, where k0 = kk*4 + 2*(lane>>4).
// C/D 16x16 f32: VGPR r -> M = r + 8*(lane>>4), N = lane&15.
// ---------------------------------------------------------------------------
__global__ void gcn_gemm_wmma_kernel(const float* __restrict__ A,   // [M,64] agg
                                     const float* __restrict__ B,   // [64,64] theta
                                     float* __restrict__ C,         // [M,64] out
                                     int M) {
    const int lane = threadIdx.x & 31;
    const int wave = threadIdx.x >> 5;
    const int wavesPerBlock = blockDim.x >> 5;

    const int colTiles = D_IN / TILE;          // 4
    int tile    = blockIdx.x * wavesPerBlock + wave;
    int rowTile = tile / colTiles;
    int colTile = tile - rowTile * colTiles;
    int rowBase = rowTile * TILE;
    int colBase = colTile * TILE;
    if (rowBase >= M) return;

    const int half = lane >> 4;                // 0: K{0,1} / M rows 0-7 of D
    const int sub  = lane & 15;                // row (A) / col (B,D) index

    v8f acc = {};
#pragma unroll
    for (int kk = 0; kk < D_IN / 4; ++kk) {
        int k0 = kk * 4 + half * 2;

        // A fragment: contiguous float2
        const float* ap = A + (size_t)(rowBase + sub) * D_IN + k0;
        v2f a;
        a.x = ap[0];
        a.y = ap[1];

        // B fragment: theta rows k0, k0+1 at column (colBase + sub)
        const float* bp = B + (size_t)k0 * D_IN + colBase + sub;
        v2f b;
        b.x = bp[0];
        b.y = bp[D_IN];

        // D = A x B + C   (fp32 WMMA, full precision vs reference)
        acc = __builtin_amdgcn_wmma_f32_16x16x4_f32(
            /*neg_a=*/false, a, /*neg_b=*/false, b,
            /*c_mod=*/(short)0, acc, /*reuse_a=*/false, /*reuse_b=*/false);
    }

    // Store D tile
#pragma unroll
    for (int r = 0; r < 8; ++r) {
        int m = rowBase + r + 8 * half;
        C[(size_t)m * D_IN + colBase + sub] = acc[r];
    }
}

// ---------------------------------------------------------------------------
// Launch: inputs are (src, dst, w, data, theta, num_nodes) per setup_inputs().
// d_ws holds the fp32 aggregation buffer agg[N, 64].
// ---------------------------------------------------------------------------
extern "C" void kernel_launch(void* const* d_in, const int* in_sizes, int n_in,
                              void* d_out, int out_size, void* d_ws, size_t ws_size,
                              hipStream_t stream) {
    const int*   src   = (const int*)  d_in[0];
    const int*   dst   = (const int*)  d_in[1];
    const float* w     = (const float*)d_in[2];
    const float* data  = (const float*)d_in[3];
    const float* theta = (const float*)d_in[4];

    const int n_edges = in_sizes[0];
    const int n_nodes = in_sizes[3] / D_IN;

    float* agg = (float*)d_ws;                  // n_nodes * 64 floats

    // 1) zero agg
    {
        int n4 = (n_nodes * D_IN) / 4;
        int threads = 256;
        int blocks  = (n4 + threads - 1) / threads;
        if (blocks > 16384) blocks = 16384;     // grid-stride
        gcn_zero_kernel<<<blocks, threads, 0, stream>>>((float4*)agg, n4);
    }

    // 2) edge scatter: one wave32 per edge
    {
        long long total_threads = (long long)n_edges * 32;
        int threads = 256;
        long long blocks = (total_threads + threads - 1) / threads;
        gcn_scatter_kernel<<<(unsigned)blocks, threads, 0, stream>>>(
            src, dst, w, data, agg, n_edges);
    }

    // 3) WMMA projection: one wave per 16x16 tile
    {
        int tiles = (n_nodes / TILE) * (D_IN / TILE);   // 6250 * 4 = 25000
        int threads = 256;                              // 8 waves/block
        int wavesPerBlock = threads / 32;
        int blocks = (tiles + wavesPerBlock - 1) / wavesPerBlock;
        gcn_gemm_wmma_kernel<<<blocks, threads, 0, stream>>>(
            agg, theta, (float*)d_out, n_nodes);
    }
}